// TrilinearInterpolation_28200755265787
// MI455X (gfx1250) — compile-verified
//
#include <hip/hip_runtime.h>

// ---------------------------------------------------------------------------
// Trilinear 3D-LUT application (32x3x512x512 image, 3x33^3 LUT), gfx1250.
// Memory-bound: ~201 MB compulsory HBM traffic -> ~8.6us floor @ 23.3 TB/s.
// Strategy: one LUT channel plane (143.7 KB) resident in WGP LDS (320 KB),
// staged with async global->LDS b128 DMA (ASYNCcnt path); grid.y = channel;
// NT stores keep the L2-resident image from being evicted across passes.
// ---------------------------------------------------------------------------

#define LUT_D     33
#define LUT_ELEMS (LUT_D * LUT_D * LUT_D)   // 35937 floats = 143748 B
#define LUT_TOTAL (3 * LUT_ELEMS)           // 107811 floats
#define PLANE     (512 * 512)               // 262144 pixels per (b,c) plane
#define NPIX      (32 * PLANE)              // 8388608 pixels per channel
#define TPB       256
#define GRIDX     512

typedef float v4f __attribute__((ext_vector_type(4)));
typedef int   v4i __attribute__((ext_vector_type(4)));

#if defined(__gfx1250__) && \
    __has_builtin(__builtin_amdgcn_global_load_async_to_lds_b32) && \
    __has_builtin(__builtin_amdgcn_s_wait_asynccnt)
#define USE_ASYNC_LDS 1
#else
#define USE_ASYNC_LDS 0
#endif

#if USE_ASYNC_LDS && __has_builtin(__builtin_amdgcn_global_load_async_to_lds_b128)
#define USE_ASYNC_B128 1
#else
#define USE_ASYNC_B128 0
#endif

// Output tuple element 0: the LUT, passed through unchanged.
__global__ __launch_bounds__(TPB)
void lut_copy_kernel(const float* __restrict__ lut, float* __restrict__ dst) {
  int i = blockIdx.x * TPB + threadIdx.x;
  if (i < LUT_TOTAL) dst[i] = lut[i];
}

__global__ __launch_bounds__(TPB)
void trilerp_kernel(const float* __restrict__ lut,
                    const float* __restrict__ img,
                    float* __restrict__ out) {
  // +4 pad: element i of channel c is stored at s_lut[i + c] so that the
  // 16B-aligned chunks of the global plane land on 16B LDS addresses.
  __shared__ __attribute__((aligned(16))) float s_lut[LUT_ELEMS + 4];

  const int c = blockIdx.y;                    // output channel this block computes
  const float* lc = lut + c * LUT_ELEMS;       // that channel's 33^3 plane
  const int tid = threadIdx.x;

  // ---- Stage lut[c] into LDS (element i -> s_lut[i + c]) --------------------
#if USE_ASYNC_B128
  {
    typedef __attribute__((address_space(1))) v4i as1_v4i;
    typedef __attribute__((address_space(3))) v4i as3_v4i;
    typedef __attribute__((address_space(1))) int as1_int;
    typedef __attribute__((address_space(3))) int as3_int;
    // first i with (35937*c + i) % 4 == 0  (35937 % 4 == 1)
    const int i0   = (4 - c) & 3;
    const int n4   = (LUT_ELEMS - i0) >> 2;    // aligned 16B chunks
    const int tail = i0 + (n4 << 2);           // first leftover element
    const int rem  = LUT_ELEMS - tail;         // 0..3 leftovers
    // head (< 4 elems): one b32 per low-id lane
    if (tid < i0)
      __builtin_amdgcn_global_load_async_to_lds_b32(
          (as1_int*)(lc + tid), (as3_int*)(&s_lut[tid + c]), 0, 0);
    // aligned body: 512B per wave-instruction of async DMA
    for (int j = tid; j < n4; j += TPB) {
      const int i = i0 + (j << 2);
      __builtin_amdgcn_global_load_async_to_lds_b128(
          (as1_v4i*)(lc + i), (as3_v4i*)(&s_lut[i + c]), 0, 0);
    }
    // tail (< 4 elems)
    if (tid < rem)
      __builtin_amdgcn_global_load_async_to_lds_b32(
          (as1_int*)(lc + tail + tid), (as3_int*)(&s_lut[tail + tid + c]), 0, 0);
    __builtin_amdgcn_s_wait_asynccnt(0);
  }
#elif USE_ASYNC_LDS
  {
    typedef __attribute__((address_space(1))) int as1_int;
    typedef __attribute__((address_space(3))) int as3_int;
    for (int i = tid; i < LUT_ELEMS; i += TPB)
      __builtin_amdgcn_global_load_async_to_lds_b32(
          (as1_int*)(lc + i), (as3_int*)(&s_lut[i + c]), 0, 0);
    __builtin_amdgcn_s_wait_asynccnt(0);
  }
#else
  for (int i = tid; i < LUT_ELEMS; i += TPB) s_lut[i + c] = lc[i];
#endif
  __syncthreads();

  const float* sl = s_lut + c;                 // channel-shifted LDS view

  // ---- Stream pixels: 4 per lane via b128 loads/stores ----------------------
  const float s = 32.0f, smax = 31.0f;
  const int stride = GRIDX * TPB * 4;          // 524288 pixels per grid step

  for (int q = (blockIdx.x * TPB + tid) * 4; q < NPIX; q += stride) {
    const int bi = q >> 18;                    // batch index (q / PLANE)
    const int p  = q & (PLANE - 1);            // pixel within plane
    const float* pin = img + (size_t)(bi * 3) * PLANE + p;

    v4f red = *(const v4f*)(pin);
    v4f grn = *(const v4f*)(pin + PLANE);
    v4f blu = *(const v4f*)(pin + 2 * PLANE);

    // hint img 2 planes (stride floats) past each of the three loads (speculative prefetch,
    // translation failures are silently dropped -> safe near buffer end)
    __builtin_prefetch(pin + stride, 0, 3);
    __builtin_prefetch(pin + PLANE + stride, 0, 3);
    __builtin_prefetch(pin + 2 * PLANE + stride, 0, 3);

    v4f o;
#pragma unroll
    for (int k = 0; k < 4; ++k) {
      float x = fminf(fmaxf(red[k] * s, 0.0f), s);
      float y = fminf(fmaxf(grn[k] * s, 0.0f), s);
      float z = fminf(fmaxf(blu[k] * s, 0.0f), s);
      float xf = fminf(floorf(x), smax);       // floor >= 0 already since x >= 0
      float yf = fminf(floorf(y), smax);
      float zf = fminf(floorf(z), smax);
      float wx = x - xf, wy = y - yf, wz = z - zf;

      int base = (int)zf * (LUT_D * LUT_D) + (int)yf * LUT_D + (int)xf;

      // 8 corners from LDS; x-adjacent pairs fuse to ds_load_2addr_b32
      float c000 = sl[base];
      float c001 = sl[base + 1];
      float c010 = sl[base + LUT_D];
      float c011 = sl[base + LUT_D + 1];
      float c100 = sl[base + LUT_D * LUT_D];
      float c101 = sl[base + LUT_D * LUT_D + 1];
      float c110 = sl[base + LUT_D * LUT_D + LUT_D];
      float c111 = sl[base + LUT_D * LUT_D + LUT_D + 1];

      float a00 = c000 + wx * (c001 - c000);
      float a01 = c010 + wx * (c011 - c010);
      float a10 = c100 + wx * (c101 - c100);
      float a11 = c110 + wx * (c111 - c110);
      float a0  = a00 + wy * (a01 - a00);
      float a1  = a10 + wy * (a11 - a10);
      o[k] = a0 + wz * (a1 - a0);
    }

    // NT store: don't let 100 MB of output evict the L2-resident image,
    // which the other two channel passes re-read.
    float* pout = out + (size_t)(bi * 3 + c) * PLANE + p;
    __builtin_nontemporal_store(o, (v4f*)pout);
  }
}

extern "C" void kernel_launch(void* const* d_in, const int* in_sizes, int n_in,
                              void* d_out, int out_size, void* d_ws, size_t ws_size,
                              hipStream_t stream) {
  (void)in_sizes; (void)n_in; (void)out_size; (void)d_ws; (void)ws_size;

  const float* lut = (const float*)d_in[0];   // (3, 33, 33, 33) f32
  const float* img = (const float*)d_in[1];   // (32, 3, 512, 512) f32
  float* out = (float*)d_out;                 // [lut | out] concatenated

  // tuple element 0: lut passthrough
  lut_copy_kernel<<<(LUT_TOTAL + TPB - 1) / TPB, TPB, 0, stream>>>(lut, out);

  // tuple element 1: interpolated image, one grid.y slice per channel
  dim3 grid(GRIDX, 3, 1);
  trilerp_kernel<<<grid, TPB, 0, stream>>>(lut, img, out + LUT_TOTAL);
}